// LambdaRankLoss_6940667150369
// MI455X (gfx1250) — compile-verified
//
#include <hip/hip_runtime.h>
#include <stdint.h>

#define MAX_N 8192
#define SIGMA 1.0f

typedef __attribute__((ext_vector_type(2))) float v2f;
typedef __attribute__((ext_vector_type(8))) float v8f;

// Faithful per-pair term. Note delta_ndcg == 0.0f exactly in the reference
// (zeros_like(ti) * (ideal_dcg/ideal_dcg)); IEEE semantics keep the sigmoid
// pipeline live because lam = x*0.0f may be -0.0.
__device__ __forceinline__ float pair_term(const float* sp, const float* st,
                                           int i, int j) {
    float d   = sp[i] - sp[j];
    float ti  = st[i], tj = st[j];
    float sij = (ti > tj) ? 1.0f : ((ti < tj) ? -1.0f : 0.0f);
    float pij = 1.0f / (1.0f + __expf(-SIGMA * d));
    float lam = ((1.0f - pij) * sij) * 0.0f;   // delta_ndcg == 0
    return (sij != 0.0f) ? (lam * d) : 0.0f;
}

__global__ void __launch_bounds__(256)
lambdarank_partial(const float* __restrict__ pred,
                   const float* __restrict__ targ,
                   const int*   __restrict__ pairs,
                   int n, int num_pairs,
                   float* __restrict__ partial) {
    __shared__ float s_pred[MAX_N];
    __shared__ float s_targ[MAX_N];
    __shared__ float s_wsum[8];

    const int tid = threadIdx.x;

    // ---- Stage gather tables into LDS via async global->LDS DMA path ----
    {
        uint32_t lp = (uint32_t)(uintptr_t)&s_pred[0];
        uint32_t lt = (uint32_t)(uintptr_t)&s_targ[0];
        for (int e = tid * 4; e < n; e += (int)blockDim.x * 4) {
            uint64_t gp = (uint64_t)(uintptr_t)(pred + e);
            uint64_t gt = (uint64_t)(uintptr_t)(targ + e);
            uint32_t lo = (uint32_t)e * 4u;
            asm volatile("global_load_async_to_lds_b128 %0, %1, off"
                         :: "v"(lp + lo), "v"(gp) : "memory");
            asm volatile("global_load_async_to_lds_b128 %0, %1, off"
                         :: "v"(lt + lo), "v"(gt) : "memory");
        }
        asm volatile("s_wait_asynccnt 0" ::: "memory");
    }
    __syncthreads();

    // ---- Stream pair_idx (HBM-bound: ~80 MB), gather from LDS ----
    const int4* p4   = (const int4*)pairs;
    const int npair2 = num_pairs >> 1;
    const int gid    = blockIdx.x * blockDim.x + tid;
    const int stride = gridDim.x * blockDim.x;

    float acc = 0.0f;
    for (int k = gid; k < npair2; k += stride) {
        int4 p = p4[k];
        acc += pair_term(s_pred, s_targ, p.x, p.y);
        acc += pair_term(s_pred, s_targ, p.z, p.w);
    }
    if ((num_pairs & 1) && gid == 0) {
        int i = pairs[2 * (num_pairs - 1) + 0];
        int j = pairs[2 * (num_pairs - 1) + 1];
        acc += pair_term(s_pred, s_targ, i, j);
    }

    // ---- Block reduction (wave32 tree + LDS) ----
    for (int o = 16; o > 0; o >>= 1)
        acc += __shfl_down(acc, o, 32);
    if ((tid & 31) == 0) s_wsum[tid >> 5] = acc;
    __syncthreads();
    if (tid == 0) {
        float b = 0.0f;
        const int nw = blockDim.x >> 5;
        for (int w = 0; w < nw; ++w) b += s_wsum[w];
        partial[blockIdx.x] = b;
    }
}

// One wave32. WMMA-based reduction of block partials:
// A (16x4 f32) holds the 32 lane partials (A.y = 0), B = ones => D rows are
// row-sums; summing the 8 D VGPRs per lane and folding lane^16 yields the
// grand total on every lane of the low half.
__global__ void __launch_bounds__(32)
lambdarank_finalize(const float* __restrict__ partial, int nparts,
                    int num_pairs, float* __restrict__ out) {
    const int lane = threadIdx.x;

    // Vectorized, pipelined partial gather: float4 per lane, 4 independent
    // accumulators, unrolled so several loads are outstanding at once
    // (avoids the serialized load->wait->add chain).
    float s0 = 0.0f, s1 = 0.0f, s2 = 0.0f, s3 = 0.0f;
    const float4* p4 = (const float4*)partial;
    const int n4 = nparts >> 2;
#pragma unroll 4
    for (int k = lane; k < n4; k += 32) {
        float4 v = p4[k];
        s0 += v.x; s1 += v.y; s2 += v.z; s3 += v.w;
    }
    for (int k = (n4 << 2) + lane; k < nparts; k += 32)
        s0 += partial[k];
    float s = (s0 + s1) + (s2 + s3);

    v2f a = {s, 0.0f};
    v2f b = {1.0f, 1.0f};
    v8f c = {};
    c = __builtin_amdgcn_wmma_f32_16x16x4_f32(
        /*neg_a=*/false, a, /*neg_b=*/false, b,
        /*c_mod=*/(short)0, c, /*reuse_a=*/false, /*reuse_b=*/false);

    float col = c[0] + c[1] + c[2] + c[3] + c[4] + c[5] + c[6] + c[7];
    col += __shfl_xor(col, 16, 32);

    if (lane == 0) out[0] = -(col / (float)num_pairs);
}

extern "C" void kernel_launch(void* const* d_in, const int* in_sizes, int n_in,
                              void* d_out, int out_size, void* d_ws, size_t ws_size,
                              hipStream_t stream) {
    const float* pred  = (const float*)d_in[0];
    const float* targ  = (const float*)d_in[1];
    const int*   pairs = (const int*)d_in[2];
    const int n         = in_sizes[0];
    const int num_pairs = in_sizes[2] / 2;

    float* partials = (float*)d_ws;
    size_t cap = ws_size / sizeof(float);
    int nblocks = 1024;
    if ((size_t)nblocks > cap) nblocks = (int)cap;
    if (nblocks < 1) nblocks = 1;

    hipLaunchKernelGGL(lambdarank_partial, dim3(nblocks), dim3(256), 0, stream,
                       pred, targ, pairs, n, num_pairs, partials);
    hipLaunchKernelGGL(lambdarank_finalize, dim3(1), dim3(32), 0, stream,
                       partials, nblocks, num_pairs, (float*)d_out);
}